// OrthogonalityLoss_16492674417376
// MI455X (gfx1250) — compile-verified
//
#include <hip/hip_runtime.h>

typedef __attribute__((ext_vector_type(2))) float v2f;
typedef __attribute__((ext_vector_type(8))) float v8f;

#define B_ 64
#define M_ 16
#define N_ 64
#define D_ 1024
#define WAVES_PER_BLOCK 8
#define NWAVES (B_ * N_) /* 4096, one wave per (b,n) */

// Main kernel: one wave32 per (b,n). Coalesced float4 streaming of the 16
// tgts rows; per-lane partial dot sums; WMMA (A=ones) cross-lane reduction.
__global__ void __launch_bounds__(WAVES_PER_BLOCK * 32)
ortho_loss_main(const float* __restrict__ src,
                const float* __restrict__ tgts,
                float* __restrict__ out,
                float* __restrict__ ws,
                int use_ws) {
  __shared__ float lds[WAVES_PER_BLOCK][M_][32];

  const int gwave = (int)((blockIdx.x * blockDim.x + threadIdx.x) >> 5);
  const int lane  = (int)(threadIdx.x & 31);
  const int w     = (int)(threadIdx.x >> 5);
  const int b     = gwave >> 6;   // / N_
  const int n     = gwave & 63;   // % N_

  const float* srcRow = src  + ((size_t)b * N_ + n) * D_;
  const float* tgtRow = tgts + (((size_t)b * M_) * N_ + n) * D_; // + m*N_*D_

  float acc[M_];
#pragma unroll
  for (int m = 0; m < M_; ++m) acc[m] = 0.0f;

  // Lane l covers d = l*4 + 128*i  (8 iterations). Fully coalesced b128 loads.
  for (int d0 = lane * 4; d0 < D_; d0 += 128) {
    float4 s = *(const float4*)(srcRow + d0);
#pragma unroll
    for (int m = 0; m < M_; ++m) {
      float4 t = *(const float4*)(tgtRow + (size_t)m * (N_ * D_) + d0);
      acc[m] = fmaf(s.x, t.x, acc[m]);
      acc[m] = fmaf(s.y, t.y, acc[m]);
      acc[m] = fmaf(s.z, t.z, acc[m]);
      acc[m] = fmaf(s.w, t.w, acc[m]);
    }
  }

  // Transpose partials through LDS so lane j holds column m=j of the
  // 16x32 (m x lane) partial matrix -> WMMA B-matrix layout.
#pragma unroll
  for (int m = 0; m < M_; ++m) lds[w][m][lane] = acc[m];
  __syncthreads();

  // C += ones(16x4) x B_c, B_c[k][j] = partial[m=j][lane=4c+k].
  // B layout (f32 4x16): VGPR0 = {K0 | K2}, VGPR1 = {K1 | K3}, columns
  // striped across lanes within each half. A=ones makes the result
  // invariant to any K-slot permutation.
  const int half = lane >> 4;
  const int j    = lane & 15;
  v8f c = {};
  v2f a;
  a.x = 1.0f;
  a.y = 1.0f;
#pragma unroll
  for (int ck = 0; ck < 8; ++ck) {
    const int bl = ck * 4 + half * 2;
    v2f bm;
    bm.x = lds[w][j][bl];
    bm.y = lds[w][j][bl + 1];
    // 8 args: (neg_a, A, neg_b, B, c_mod, C, reuse_a, reuse_b)
    c = __builtin_amdgcn_wmma_f32_16x16x4_f32(
        false, a, false, bm, (short)0, c, false, false);
  }

  // Every lane now holds dot[m = lane & 15] in c[0] (all C rows equal).
  float av = __builtin_fabsf(c[0]);
  // Sum within each 16-lane group -> lane 0 has sum_m |dot_m| for this (b,n).
  av += __shfl_xor(av, 1);
  av += __shfl_xor(av, 2);
  av += __shfl_xor(av, 4);
  av += __shfl_xor(av, 8);

  if (lane == 0) {
    if (use_ws) ws[gwave] = av;
    else        atomicAdd(out, av);
  }
}

__global__ void zero_out_kernel(float* out) { out[0] = 0.0f; }

// Deterministic final reduction of the 4096 per-wave sums.
__global__ void reduce_ws_kernel(const float* __restrict__ ws,
                                 float* __restrict__ out) {
  __shared__ float sm[256];
  float s = 0.0f;
  for (int i = (int)threadIdx.x; i < NWAVES; i += 256) s += ws[i];
  sm[threadIdx.x] = s;
  __syncthreads();
  for (int st = 128; st > 0; st >>= 1) {
    if ((int)threadIdx.x < st) sm[threadIdx.x] += sm[threadIdx.x + st];
    __syncthreads();
  }
  if (threadIdx.x == 0) out[0] = sm[0];
}

extern "C" void kernel_launch(void* const* d_in, const int* in_sizes, int n_in,
                              void* d_out, int out_size, void* d_ws, size_t ws_size,
                              hipStream_t stream) {
  (void)in_sizes; (void)n_in; (void)out_size;
  const float* src  = (const float*)d_in[0];
  const float* tgts = (const float*)d_in[1];
  float* out = (float*)d_out;
  float* ws  = (float*)d_ws;

  const int use_ws =
      (ws != nullptr && ws_size >= (size_t)NWAVES * sizeof(float)) ? 1 : 0;

  const int blocks = NWAVES / WAVES_PER_BLOCK; // 512 blocks x 256 threads

  if (!use_ws) {
    zero_out_kernel<<<1, 1, 0, stream>>>(out);
  }
  ortho_loss_main<<<blocks, WAVES_PER_BLOCK * 32, 0, stream>>>(
      src, tgts, out, ws, use_ws);
  if (use_ws) {
    reduce_ws_kernel<<<1, 256, 0, stream>>>(ws, out);
  }
}